// VQLayer_42485816492290
// MI455X (gfx1250) — compile-verified
//
#include <hip/hip_runtime.h>
#include <hip/hip_bf16.h>

// Problem constants (match reference)
#define K_CB   1024
#define D_CB   64
#define N_X    131072
#define GAMMA_ 0.99f
#define ALPHA_ 1e-9f
#define BETA_  0.25f

#define E_STRIDE 68   // padded LDS row stride (floats): 16B-aligned rows, conflict-free banks

typedef float v2f __attribute__((ext_vector_type(2)));
typedef float v8f __attribute__((ext_vector_type(8)));

// ---------------- kernel 0: zero workspace accumulators ----------------
__global__ void vq_zero_ws(float* __restrict__ ws, int n) {
    int i = blockIdx.x * blockDim.x + threadIdx.x;
    if (i < n) ws[i] = 0.0f;
}

// ---------------- kernel 1: distances + argmin + Z gather via fp32 WMMA ----------------
// 256 threads = 8 waves; each wave owns TWO 16-row X slabs (32 rows); block covers 256 rows.
// All of E (256KB) + ||e||^2 staged in LDS once per block; B fragments reused by both tiles.
__global__ void __launch_bounds__(256)
vq_dist_argmin(const float* __restrict__ X,
               const float* __restrict__ E,
               float* __restrict__ out_z,
               float* __restrict__ out_argmin_f,
               float* __restrict__ out_mindist,
               int*   __restrict__ ws_argmin_i,
               float* __restrict__ ws_counts,
               float* __restrict__ ws_loss) {
    extern __shared__ float smem[];
    float* sE   = smem;                      // K_CB * E_STRIDE floats
    float* sEsq = smem + K_CB * E_STRIDE;    // K_CB floats
    int*   sIdx = (int*)(sEsq + K_CB);       // 256 ints: per-row winning code

    const int tid = threadIdx.x;

    // Cooperatively stage E into LDS and compute ||e||^2 on the way in.
    for (int c = tid; c < K_CB; c += 256) {
        const float4* src = (const float4*)(E + c * D_CB);
        float4* dst = (float4*)(sE + c * E_STRIDE);
        float s = 0.0f;
        #pragma unroll
        for (int q = 0; q < D_CB / 4; ++q) {
            float4 v = src[q];
            dst[q] = v;
            s += v.x * v.x + v.y * v.y + v.z * v.z + v.w * v.w;
        }
        sEsq[c] = s;
    }
    __syncthreads();

    const int wave = tid >> 5;        // 8 waves (wave32)
    const int lane = tid & 31;
    const int half = lane >> 4;       // lane 0-15 vs 16-31
    const int m    = lane & 15;
    const int row0 = blockIdx.x * 256 + wave * 32;

    // A fragments (16x4 fp32, ISA 7.12.2): lane m = row, half selects K pair;
    // slice s covers K-dim (=D) [4s, 4s+4). float2 = {K even, K odd}. Two row tiles.
    v2f a[2][16];
    float xp[2] = {0.0f, 0.0f};
    #pragma unroll
    for (int t = 0; t < 2; ++t) {
        #pragma unroll
        for (int s = 0; s < 16; ++s) {
            const float2 v = *(const float2*)(X + (row0 + 16 * t + m) * D_CB + 4 * s + 2 * half);
            a[t][s][0] = v.x; a[t][s][1] = v.y;
            xp[t] += v.x * v.x + v.y * v.y;
        }
    }
    // lane r (and r+16) hold ||x||^2 of row (tile base + (r&15))
    float xsq[2];
    xsq[0] = xp[0] + __shfl_xor(xp[0], 16, 32);
    xsq[1] = xp[1] + __shfl_xor(xp[1], 16, 32);

    float minv[2][8];
    int   mini[2][8];
    #pragma unroll
    for (int t = 0; t < 2; ++t)
        #pragma unroll
        for (int r = 0; r < 8; ++r) { minv[t][r] = 3.4e38f; mini[t][r] = 0; }

    // Sweep all 1024 codes, 16 at a time; each B fragment feeds two WMMA chains.
    for (int cb = 0; cb < K_CB; cb += 16) {
        const float esq = sEsq[cb + m];       // this lane's code column
        v8f acc0 = {}, acc1 = {};
        #pragma unroll
        for (int s = 0; s < 16; ++s) {
            // B fragment (4x16 fp32): lane m = code column, half selects K pair.
            v2f b = *(const v2f*)(sE + (cb + m) * E_STRIDE + 4 * s + 2 * half);
            acc0 = __builtin_amdgcn_wmma_f32_16x16x4_f32(
                false, a[0][s], false, b, (short)0, acc0, false, false);
            acc1 = __builtin_amdgcn_wmma_f32_16x16x4_f32(
                false, a[1][s], false, b, (short)0, acc1, false, false);
        }
        // C layout: lanes 0-15 -> rows r, lanes 16-31 -> rows r+8; col = m.
        // ||x||^2 is row-constant -> compare score = ||e||^2 - 2*dot only.
        #pragma unroll
        for (int r = 0; r < 8; ++r) {
            float s0 = fmaf(-2.0f, acc0[r], esq);
            if (s0 < minv[0][r]) { minv[0][r] = s0; mini[0][r] = cb + m; }
            float s1 = fmaf(-2.0f, acc1[r], esq);
            if (s1 < minv[1][r]) { minv[1][r] = s1; mini[1][r] = cb + m; }
        }
    }

    // Cross-lane argmin within each 16-lane half (codes are striped across lanes).
    float lsum = 0.0f;
    #pragma unroll
    for (int t = 0; t < 2; ++t) {
        #pragma unroll
        for (int r = 0; r < 8; ++r) {
            float mv = minv[t][r]; int mi = mini[t][r];
            #pragma unroll
            for (int off = 1; off < 16; off <<= 1) {
                float ov = __shfl_xor(mv, off, 32);
                int   oi = __shfl_xor(mi, off, 32);
                if (ov < mv || (ov == mv && oi < mi)) { mv = ov; mi = oi; }
            }
            const int rloc = 16 * t + r + 8 * half;       // row within wave's 32
            const int row  = row0 + rloc;
            const float xr = __shfl(xsq[t], r + 8 * half, 32);   // all lanes execute
            const float md = mv + xr;                             // true min squared distance
            if (m == 0) {
                out_argmin_f[row] = (float)mi;
                ws_argmin_i[row]  = mi;
                out_mindist[row]  = md;
                sIdx[wave * 32 + rloc] = mi;
                atomicAdd(&ws_counts[mi], 1.0f);
                lsum += md;
            }
        }
    }
    if (m == 0) atomicAdd(ws_loss, lsum);
    __syncthreads();   // sIdx visible to whole block (cheap; also orders LDS writes)

    // Fused gather: Z[row] = E[argmin[row]] straight out of LDS.
    // Per row: 32 lanes x float2 = 64 floats; banks 2*lane -> conflict-free;
    // global stores are fully coalesced 256B lines.
    #pragma unroll 4
    for (int rr = 0; rr < 32; ++rr) {
        const int idx = sIdx[wave * 32 + rr];
        const v2f z = *(const v2f*)(sE + idx * E_STRIDE + 2 * lane);
        float2 zo; zo.x = z[0]; zo.y = z[1];
        *(float2*)(out_z + (size_t)(row0 + rr) * D_CB + 2 * lane) = zo;
    }
}

// ---------------- kernel 2: sums[idx] += X (segment sum scatter) ----------------
__global__ void __launch_bounds__(256)
vq_scatter(const float* __restrict__ X,
           const int* __restrict__ argmin_i,
           float* __restrict__ ws_sums) {
    const int gid = blockIdx.x * 256 + threadIdx.x;   // over N*D
    const int row = gid >> 6;
    const int d   = gid & 63;
    const int idx = argmin_i[row];
    atomicAdd(&ws_sums[idx * D_CB + d], X[gid]);
}

// ---------------- kernel 3: EMA finalize (single 1024-thread block) ----------------
__global__ void __launch_bounds__(1024)
vq_finalize(const float* __restrict__ cluster_sizes,
            const float* __restrict__ moving_avg,
            const float* __restrict__ ws_counts,
            const float* __restrict__ ws_sums,
            const float* __restrict__ ws_loss,
            float* __restrict__ out_loss,
            float* __restrict__ out_Enew,
            float* __restrict__ out_cs,
            float* __restrict__ out_ma) {
    __shared__ float red[1024];
    const int k = threadIdx.x;
    const float csr = GAMMA_ * cluster_sizes[k] + (1.0f - GAMMA_) * ws_counts[k];
    red[k] = csr;
    __syncthreads();
    for (int s = 512; s > 0; s >>= 1) {
        if (k < s) red[k] += red[k + s];
        __syncthreads();
    }
    const float total = red[0];
    const float cs = (csr + ALPHA_) / (1.0f + ALPHA_ * (float)K_CB / total);
    out_cs[k] = cs;
    const float inv = 1.0f / cs;
    #pragma unroll 4
    for (int d = 0; d < D_CB; ++d) {
        const float ma = GAMMA_ * moving_avg[k * D_CB + d]
                       + (1.0f - GAMMA_) * ws_sums[k * D_CB + d];
        out_ma[k * D_CB + d]   = ma;
        out_Enew[k * D_CB + d] = ma * inv;
    }
    if (k == 0) out_loss[0] = BETA_ * ws_loss[0] / (float)N_X;
}

extern "C" void kernel_launch(void* const* d_in, const int* in_sizes, int n_in,
                              void* d_out, int out_size, void* d_ws, size_t ws_size,
                              hipStream_t stream) {
    const float* X  = (const float*)d_in[0];   // [N, D]
    const float* E  = (const float*)d_in[1];   // [K, D]
    const float* CS = (const float*)d_in[2];   // [K]
    const float* MA = (const float*)d_in[3];   // [K, D]

    // Output layout (floats, tuple order): Z[N*D], loss[1], argmin[N], min_dist[N],
    // E_new[K*D], cs[K], ma[K*D]
    float* out       = (float*)d_out;
    float* out_z     = out;
    float* out_loss  = out + (size_t)N_X * D_CB;
    float* out_arg   = out_loss + 1;
    float* out_md    = out_arg + N_X;
    float* out_Enew  = out_md + N_X;
    float* out_cs    = out_Enew + (size_t)K_CB * D_CB;
    float* out_ma    = out_cs + K_CB;

    // Workspace: counts[K], sums[K*D], loss[1], argmin_i[N]
    float* ws_counts = (float*)d_ws;
    float* ws_sums   = ws_counts + K_CB;
    float* ws_loss   = ws_sums + (size_t)K_CB * D_CB;
    int*   ws_argmin = (int*)(ws_loss + 1);

    const int n_zero = K_CB + K_CB * D_CB + 1;
    vq_zero_ws<<<(n_zero + 255) / 256, 256, 0, stream>>>((float*)d_ws, n_zero);

    const size_t smem_bytes =
        (size_t)(K_CB * E_STRIDE + K_CB) * sizeof(float) + 256 * sizeof(int); // ~280 KB < 320 KB WGP LDS
    hipFuncSetAttribute((const void*)vq_dist_argmin,
                        hipFuncAttributeMaxDynamicSharedMemorySize, (int)smem_bytes);
    vq_dist_argmin<<<N_X / 256, 256, smem_bytes, stream>>>(
        X, E, out_z, out_arg, out_md, ws_argmin, ws_counts, ws_loss);

    vq_scatter<<<(N_X * D_CB) / 256, 256, 0, stream>>>(X, ws_argmin, ws_sums);

    vq_finalize<<<1, 1024, 0, stream>>>(
        CS, MA, ws_counts, ws_sums, ws_loss, out_loss, out_Enew, out_cs, out_ma);
}